// ConditionalAttentionPooling_22067541967689
// MI455X (gfx1250) — compile-verified
//
#include <hip/hip_runtime.h>
#include <hip/hip_bf16.h>

#define BDIM 64
#define SDIM 16
#define JDIM 256
#define QD   512
#define ID   512
#define HD   256

typedef float v2f __attribute__((ext_vector_type(2)));
typedef float v8f __attribute__((ext_vector_type(8)));

// ---------------------------------------------------------------------------
// f32 WMMA GEMM: C[M,N] = A[M,K] @ B  (BT=1: B stored [N,K]; BT=0: B stored [K,N])
// One wave per 16x16 tile, V_WMMA_F32_16X16X4_F32 chained over K.
// (Operand layouts per ISA 7.12.2; confirmed lowering in round 1 disasm.)
// ---------------------------------------------------------------------------
template <bool BT>
__global__ __launch_bounds__(32) void wmma_gemm_f32(
    const float* __restrict__ A, const float* __restrict__ Bm,
    float* __restrict__ C, int M, int N, int K) {
  const int ntn  = N >> 4;
  const int m0   = (blockIdx.x / ntn) << 4;
  const int n0   = (blockIdx.x % ntn) << 4;
  const int lane = threadIdx.x & 31;
  const int half = lane >> 4;
  const int lo   = lane & 15;

  v8f c = {};
  const float* arow = A + (size_t)(m0 + lo) * K + 2 * half;
  const float* brow = BT ? (Bm + (size_t)(n0 + lo) * K + 2 * half) : Bm;

  for (int k = 0; k < K; k += 4) {
    v2f a, b;
    a.x = arow[k];
    a.y = arow[k + 1];
    if (BT) {
      b.x = brow[k];
      b.y = brow[k + 1];
    } else {
      b.x = Bm[(size_t)(k + 2 * half) * N + n0 + lo];
      b.y = Bm[(size_t)(k + 2 * half + 1) * N + n0 + lo];
    }
    c = __builtin_amdgcn_wmma_f32_16x16x4_f32(false, a, false, b, (short)0, c,
                                              false, false);
  }
  float* crow = C + n0 + lo;
#pragma unroll
  for (int r = 0; r < 8; ++r) crow[(size_t)(m0 + r + 8 * half) * N] = c[r];
}

// ---------------------------------------------------------------------------
// Async copy: 4 x GLOBAL_LOAD_ASYNC_TO_LDS_B128 per thread (one 8x512 f32
// chunk = 16 KB per workgroup). IOFFSET applies to both LDS and global sides.
// Tracked by ASYNCcnt (4 per wave per chunk).
// ---------------------------------------------------------------------------
__device__ __forceinline__ void async_copy_chunk(const float* gsrc,
                                                 unsigned lds) {
  asm volatile(
      "global_load_async_to_lds_b128 %0, %1, off\n\t"
      "global_load_async_to_lds_b128 %0, %1, off offset:4096\n\t"
      "global_load_async_to_lds_b128 %0, %1, off offset:8192\n\t"
      "global_load_async_to_lds_b128 %0, %1, off offset:12288" ::"v"(lds),
      "v"(gsrc)
      : "memory");
}

// ---------------------------------------------------------------------------
// Streaming fused masked-softmax pooling: one workgroup (256 thr) per (b,s).
// Double-buffered async-to-LDS chunks (8 rows), online softmax over J=256,
// acc[512] in registers (2 per thread). wx[bs,i] = softmax(X@qk) . X
// ---------------------------------------------------------------------------
#define CJ  8
#define NCH (JDIM / CJ)  // 32 chunks

__global__ __launch_bounds__(256) void stream_attn_kernel(
    const float* __restrict__ X,             // [B,S,J,ID]
    const unsigned char* __restrict__ mask,  // [B,S,J] (jax bool = 1 byte)
    const float* __restrict__ qk,            // [B, ID]
    float* __restrict__ wx) {                // [B*S, ID]
  __shared__ float qk_s[ID];
  __shared__ float xs[2][CJ * ID];
  __shared__ float p_s[CJ];
  __shared__ unsigned char msk_s[JDIM];

  const int t  = threadIdx.x;
  const int bs = blockIdx.x;  // b*S + s
  const int b  = bs >> 4;

  qk_s[t]       = qk[b * ID + t];
  qk_s[t + 256] = qk[b * ID + t + 256];
  msk_s[t]      = mask[(size_t)bs * JDIM + t];
  // dummy init: establish xs as written before async engine takes over
  xs[0][t] = 0.f;
  xs[1][t] = 0.f;
  asm volatile("s_wait_dscnt 0x0" ::: "memory");

  const float* Xbs = X + (size_t)bs * JDIM * ID;

  const int g = t & 31;  // lane: one wave per row
  const int r = t >> 5;  // wave id = row 0..7 within chunk

  const float NEGINF = -__builtin_inff();
  float m = NEGINF, l = 0.f, acc0 = 0.f, acc1 = 0.f;

  // per-thread async addresses: thread t copies float4 e = t + 256*q, q=0..3
  const float* gbase   = Xbs + t * 4;
  const unsigned lds0  = (unsigned)(uintptr_t)(&xs[0][0]) + t * 16;
  const unsigned lds1  = (unsigned)(uintptr_t)(&xs[1][0]) + t * 16;

  async_copy_chunk(gbase, lds0);  // chunk 0 -> buf 0

  for (int c = 0; c < NCH; ++c) {
    const int cur = c & 1;
    if (c + 1 < NCH) {
      async_copy_chunk(gbase + (size_t)(c + 1) * CJ * ID, cur ? lds0 : lds1);
      // async loads complete in order: <=4 outstanding => chunk c has landed
      asm volatile("s_wait_asynccnt 0x4" ::: "memory");
    } else {
      asm volatile("s_wait_asynccnt 0x0" ::: "memory");
    }
    __syncthreads();

    const float* xc = xs[cur];

    // scores[r] = X[r,:].qk : full-wave strided partials + shuffle reduce
    float part = 0.f;
#pragma unroll
    for (int k = 0; k < 16; ++k)
      part += xc[r * ID + g + 32 * k] * qk_s[g + 32 * k];
    part += __shfl_down(part, 16);
    part += __shfl_down(part, 8);
    part += __shfl_down(part, 4);
    part += __shfl_down(part, 2);
    part += __shfl_down(part, 1);
    if (g == 0) {
      float sc = part * 0.0625f;                 // 1/sqrt(HD=256)
      if (msk_s[c * CJ + r]) sc = NEGINF;        // mask==True -> masked out
      p_s[r] = sc;
    }
    __syncthreads();

    // online-softmax update (identical scalars recomputed by every thread)
    float mc = NEGINF;
#pragma unroll
    for (int j = 0; j < CJ; ++j) mc = fmaxf(mc, p_s[j]);
    float mn = fmaxf(m, mc);
    if (mn != NEGINF) {  // else: everything so far masked, nothing to add
      float alpha = __expf(m - mn);  // m may be -inf -> alpha = 0
      l *= alpha;
      acc0 *= alpha;
      acc1 *= alpha;
#pragma unroll
      for (int j = 0; j < CJ; ++j) {
        float pj = __expf(p_s[j] - mn);  // -inf score -> 0 weight
        l += pj;
        acc0 += pj * xc[j * ID + t];
        acc1 += pj * xc[j * ID + t + 256];
      }
      m = mn;
    }
    __syncthreads();  // everyone done with buf[cur] before it is refilled
  }

  float inv = 1.0f / l;  // l==0 (all masked) -> inf/NaN like jax softmax
  wx[(size_t)bs * ID + t]       = acc0 * inv;
  wx[(size_t)bs * ID + t + 256] = acc1 * inv;
}

// ---------------------------------------------------------------------------
extern "C" void kernel_launch(void* const* d_in, const int* in_sizes, int n_in,
                              void* d_out, int out_size, void* d_ws,
                              size_t ws_size, hipStream_t stream) {
  const float* query        = (const float*)d_in[0];
  const float* X            = (const float*)d_in[1];
  const unsigned char* mask = (const unsigned char*)d_in[2];
  const float* Wq           = (const float*)d_in[3];
  const float* Wk           = (const float*)d_in[4];
  const float* Wv           = (const float*)d_in[5];
  float* out                = (float*)d_out;

  float* Q  = (float*)d_ws;   // [64,256]
  float* qk = Q + BDIM * HD;  // [64,512]
  float* wx = qk + BDIM * ID; // [1024,512]

  // Q = query @ Wq^T   (Wq stored [HD, QD] = [N, K])
  wmma_gemm_f32<true><<<(BDIM / 16) * (HD / 16), 32, 0, stream>>>(
      query, Wq, Q, BDIM, HD, QD);
  // qk = Q @ Wk        (Wk stored [HD, ID] = [K, N])
  wmma_gemm_f32<false><<<(BDIM / 16) * (ID / 16), 32, 0, stream>>>(
      Q, Wk, qk, BDIM, ID, HD);
  // fused streaming scores+softmax+pool over 512 MB of X (memory-bound core)
  stream_attn_kernel<<<BDIM * SDIM, 256, 0, stream>>>(X, mask, qk, wx);
  // out = wx @ Wv^T    (Wv stored [HD, ID] = [N, K])
  wmma_gemm_f32<true><<<(BDIM * SDIM / 16) * (HD / 16), 32, 0, stream>>>(
      wx, Wv, out, BDIM * SDIM, HD, QD);
}